// Laplace_46720654246197
// MI455X (gfx1250) — compile-verified
//
#include <hip/hip_runtime.h>
#include <math.h>

#define TAU_MIN  1.0f
#define TAU_MAX  20.0f
#define KCONST   4
#define NTAU     100
#define ALPHA    1.0f
#define F        512
#define T        2048
#define S_TOT    (NTAU + 2*KCONST)   /* 108 */
#define CT       128                 /* t-chunk length */
#define NCHUNK   (T / CT)            /* 16 */

typedef __attribute__((ext_vector_type(2))) float v2f;
typedef __attribute__((ext_vector_type(8))) float v8f;

__device__ __forceinline__ float laplace_sval(int sidx) {
    // c = (tau_max/tau_min)^(1/(ntau-1)) - 1 ; tau_star = tau_min*(1+c)^(sidx-K); s = K/tau_star
    float c = powf(TAU_MAX / TAU_MIN, 1.0f / (float)(NTAU - 1)) - 1.0f;
    float tau_star = TAU_MIN * powf(1.0f + c, (float)(sidx - KCONST));
    return (float)KCONST / tau_star;
}

// Pass 1: per (s, chunk, f) partial scan from zero state: r = sum_j a^{CT-1-j} * decay * x[t0+j,f]
__global__ void laplace_chunk_reduce(const float* __restrict__ inp, float* __restrict__ ws1) {
    int tid  = blockIdx.x * blockDim.x + threadIdx.x;
    int f    = tid % F;
    int rem  = tid / F;
    int chunk = rem % NCHUNK;
    int sidx  = rem / NCHUNK;
    if (sidx >= S_TOT) return;

    float sv    = laplace_sval(sidx);
    float a     = expf(-sv * (TAU_MIN * ALPHA));
    float decay = (1.0f - a) / sv;

    const float* p = inp + (size_t)chunk * CT * F + f;
    float r = 0.0f;
#pragma unroll 8
    for (int j = 0; j < CT; ++j)
        r = fmaf(a, r, decay * p[(size_t)j * F]);
    ws1[((size_t)sidx * NCHUNK + chunk) * F + f] = r;
}

// Pass 2: per (s,f) carry scan across chunks: state_{c+1} = a^CT * state_c + r_c ; store carry INTO each chunk
__global__ void laplace_carry_scan(const float* __restrict__ ws1, float* __restrict__ ws2) {
    int tid  = blockIdx.x * blockDim.x + threadIdx.x;
    int f    = tid % F;
    int sidx = tid / F;
    if (sidx >= S_TOT) return;

    float sv  = laplace_sval(sidx);
    float aCT = expf(-sv * (TAU_MIN * ALPHA) * (float)CT);

    float st = 0.0f;
#pragma unroll
    for (int c2 = 0; c2 < NCHUNK; ++c2) {
        size_t idx = ((size_t)sidx * NCHUNK + c2) * F + f;
        ws2[idx] = st;                       // state entering chunk c2
        st = fmaf(aCT, st, ws1[idx]);
    }
}

// Pass 3: one wave per (s, f-tile of 16, t-chunk of 128).
// Per 16-step sub-chunk: out(16x16) = M'(16x16, lower-tri decay*a^{i-j}) @ X(16x16) + a^{i+1}*state
// computed as 4 chained V_WMMA_F32_16X16X4_F32; new state = row 15 of D.
__global__ void __launch_bounds__(32)
laplace_wmma_out(const float* __restrict__ inp, const float* __restrict__ ws2,
                 float* __restrict__ out) {
    const int lane = threadIdx.x;       // 0..31 (wave32)
    const int hi   = lane >> 4;         // lane half
    const int m    = lane & 15;         // A: M-row / B,C,D: column

    int bid   = blockIdx.x;             // [S_TOT][NCHUNK][F/16]
    int ftile = bid % (F / 16);
    int rem   = bid / (F / 16);
    int chunk = rem % NCHUNK;
    int sidx  = rem / NCHUNK;
    int f0 = ftile * 16;
    int t0 = chunk * CT;

    float sv    = laplace_sval(sidx);
    float sa    = sv * (TAU_MIN * ALPHA);      // a^p = exp(-sa*p)
    float a     = expf(-sa);
    float decay = (1.0f - a) / sv;

    // A-matrix (16x4 per WMMA): lanes = M rows; VGPR0 holds K even (lo half) / K+2 (hi half), VGPR1 the odd K.
    v2f Ablk[4];
#pragma unroll
    for (int b = 0; b < 4; ++b) {
        int k0 = 4 * b + 2 * hi;
#pragma unroll
        for (int j = 0; j < 2; ++j) {
            int k = k0 + j;
            Ablk[b][j] = (k <= m) ? decay * expf(-sa * (float)(m - k)) : 0.0f;
        }
    }

    // Carry multipliers: C/D VGPR r maps to row (r + 8*hi); factor = a^(row+1)
    float vsel[8];
#pragma unroll
    for (int r = 0; r < 8; ++r)
        vsel[r] = expf(-sa * (float)(r + 8 * hi + 1));

    // State entering this 128-chunk for column f0+m (same value in both lane halves)
    float state = ws2[((size_t)sidx * NCHUNK + chunk) * F + f0 + m];

    const float* ip = inp + (size_t)t0 * F + f0 + m;
    float*       op = out + (size_t)t0 * S_TOT * F + (size_t)sidx * F + f0 + m;
    const size_t outRowStride = (size_t)S_TOT * F;

    for (int sub = 0; sub < CT / 16; ++sub) {
        // B-matrix (4x16 per WMMA): row K striped across lanes; same even/odd K split as A.
        v2f Bblk[4];
#pragma unroll
        for (int b = 0; b < 4; ++b) {
            int k0 = 4 * b + 2 * hi;
            Bblk[b][0] = ip[(size_t)(sub * 16 + k0)     * F];
            Bblk[b][1] = ip[(size_t)(sub * 16 + k0 + 1) * F];
        }

        // Accumulator init = rank-1 carry term a^{row+1} * state[col]
        v8f acc;
#pragma unroll
        for (int r = 0; r < 8; ++r) acc[r] = vsel[r] * state;

        // K = 16 via 4 chained f32 WMMAs
        acc = __builtin_amdgcn_wmma_f32_16x16x4_f32(false, Ablk[0], false, Bblk[0], (short)0, acc, false, false);
        acc = __builtin_amdgcn_wmma_f32_16x16x4_f32(false, Ablk[1], false, Bblk[1], (short)0, acc, false, false);
        acc = __builtin_amdgcn_wmma_f32_16x16x4_f32(false, Ablk[2], false, Bblk[2], (short)0, acc, false, false);
        acc = __builtin_amdgcn_wmma_f32_16x16x4_f32(false, Ablk[3], false, Bblk[3], (short)0, acc, false, false);

        // Store tile: VGPR r -> out row (t0 + sub*16 + r + 8*hi), 16 contiguous floats per lane half
#pragma unroll
        for (int r = 0; r < 8; ++r)
            op[(size_t)(sub * 16 + r + 8 * hi) * outRowStride] = acc[r];

        // New state = D row 15 = acc[7] in lanes 16..31 (column m); broadcast to both halves
        state = __shfl(acc[7], 16 + m, 32);
    }
}

extern "C" void kernel_launch(void* const* d_in, const int* in_sizes, int n_in,
                              void* d_out, int out_size, void* d_ws, size_t ws_size,
                              hipStream_t stream) {
    (void)in_sizes; (void)n_in; (void)out_size; (void)ws_size;
    const float* inp = (const float*)d_in[0];
    float* out = (float*)d_out;
    float* ws1 = (float*)d_ws;                               // [S_TOT][NCHUNK][F] chunk reductions
    float* ws2 = ws1 + (size_t)S_TOT * NCHUNK * F;           // [S_TOT][NCHUNK][F] chunk carries

    int n1 = S_TOT * NCHUNK * F;      // 884,736 -> 3456 blocks
    laplace_chunk_reduce<<<n1 / 256, 256, 0, stream>>>(inp, ws1);

    int n2 = S_TOT * F;               // 55,296 -> 216 blocks
    laplace_carry_scan<<<n2 / 256, 256, 0, stream>>>(ws1, ws2);

    int ntile = S_TOT * NCHUNK * (F / 16);   // 55,296 single-wave workgroups
    laplace_wmma_out<<<ntile, 32, 0, stream>>>(inp, ws2, out);
}